// Model_43147241455977
// MI455X (gfx1250) — compile-verified
//
#include <hip/hip_runtime.h>
#include <math.h>

// ---------------- problem constants ----------------
#define Gn    128
#define Nn    512
#define Fh    128           // F_IN == H == 128
#define Cc    10
#define LAMB      1.0f
#define NEG_SLOPE 0.2f

// ---------------- per-graph ws arena (floats) ----------------
#define ARENA      262144L          // 1 MB per graph, 128 MB total
#define XW_OFF     0L               // 512*128
#define H_OFF      65536L           // 512*128
#define HP_OFF     131072L          // 256*128 pooled features
#define A1_OFF     163840L          // 256*256 attn adjacency (layer 2)
#define A2_OFF     229376L          // 128*128 attn adjacency (layer 3)
#define DINV_OFF   245760L          // 512
#define DEGINV_OFF 246272L          // 512
#define S_OFF      246784L          // 512
#define IDX_OFF    247296L          // 512 (ints)
#define AS_OFF     247808L          // 512
#define AD_OFF     248320L          // 512
#define XR_OFF     248832L          // 768: x1|x2|x3 readouts

typedef __attribute__((ext_vector_type(2))) float v2f;
typedef __attribute__((ext_vector_type(8))) float v8f;

#define WMMA_F32(A, B, C) \
  __builtin_amdgcn_wmma_f32_16x16x4_f32(false, (A), false, (B), (short)0, (C), false, false)

// =====================================================================
// degree / scale vectors:  dinv = rsqrt(deg+1),  deginv = 1/max(deg,1)
// wave per row, float4 lane-strided loads, shfl reduction. grid (G, n/8)
// =====================================================================
__global__ void k_deg(const float* __restrict__ adjBase, long adjGS, int n,
                      float* __restrict__ ws) {
  const int g = blockIdx.x, tid = threadIdx.x;
  const int wave = tid >> 5, lane = tid & 31;
  const int i = blockIdx.y * 8 + wave;
  const float* row = adjBase + (size_t)g * adjGS + (size_t)i * n;
  float s = 0.0f;
  for (int j = lane * 4; j < n; j += 128) {
    const float4 v = *(const float4*)&row[j];
    s += v.x + v.y + v.z + v.w;
  }
  for (int m = 16; m >= 1; m >>= 1) s += __shfl_xor(s, m, 32);
  if (lane == 0) {
    float* arena = ws + (size_t)g * ARENA;
    (arena + DINV_OFF)[i]   = rsqrtf(s + 1.0f);
    (arena + DEGINV_OFF)[i] = 1.0f / fmaxf(s, 1.0f);
  }
}

// =====================================================================
// XW = X @ W   (M x 128) * (128 x 128), WMMA f32 16x16x4
// block = 256 = 8 waves; 32-row macro-tile; wave -> TWO 16x16 tiles
// sharing the B fragment (2 independent WMMA chains). grid (G, M/32)
// =====================================================================
__global__ void k_xw(const float* __restrict__ Xbase, long xGS,
                     const float* __restrict__ W,
                     float* __restrict__ ws, int M) {
  __shared__ float AT[32][132];           // stride 132: banks 4*tl, no conflicts
  const int g    = blockIdx.x;
  const int row0 = blockIdx.y * 32;
  const int tid  = threadIdx.x;
  const int wave = tid >> 5, lane = tid & 31;
  const int half = lane >> 4, tl = lane & 15;
  const int cb   = wave * 16;

  const float* Xg = Xbase + (size_t)g * xGS + (size_t)row0 * Fh;
#pragma unroll
  for (int p = 0; p < 4; ++p) {                       // 32x128 = 1024 float4
    const int e = tid + 256 * p;
    const int r = e >> 5, c4 = (e & 31) << 2;
    *(float4*)&AT[r][c4] = *(const float4*)&Xg[(size_t)r * Fh + c4];
  }
  __syncthreads();

  v8f acc0 = {}, acc1 = {};
  for (int k0 = 0; k0 < Fh; k0 += 4) {
    v2f a0, a1, b;
    a0.x = AT[tl][k0 + 2 * half];      a0.y = AT[tl][k0 + 2 * half + 1];
    a1.x = AT[16 + tl][k0 + 2 * half]; a1.y = AT[16 + tl][k0 + 2 * half + 1];
    b.x  = W[(size_t)(k0 + 2 * half) * Fh + cb + tl];
    b.y  = W[(size_t)(k0 + 2 * half + 1) * Fh + cb + tl];
    acc0 = WMMA_F32(a0, b, acc0);
    acc1 = WMMA_F32(a1, b, acc1);
  }
  float* XW = ws + (size_t)g * ARENA + XW_OFF;
#pragma unroll
  for (int r = 0; r < 8; ++r) {
    XW[(size_t)(row0 + r + 8 * half) * Fh + cb + tl]      = acc0[r];
    XW[(size_t)(row0 + 16 + r + 8 * half) * Fh + cb + tl] = acc1[r];
  }
}

// =====================================================================
// agg = adj @ B   (M x M) * (M x 128), WMMA f32, LDS tiles (KT=64),
// 32-row macro-tile, dual accumulators, register double-buffered staging.
// mode 0 (GCN):  B = dinv[k]*XW[k][:];  H = relu(dinv_i*agg + dinv_i^2*xw + b)
// mode 1 (info): B = H[k][:];           S[i] = sum_f |H[i,f] - agg*deginv_i|
// grid (G, M/32)
// =====================================================================
#define KT 64
__global__ void k_agg(const float* __restrict__ adjBase, long adjGS,
                      const float* __restrict__ bias,
                      float* __restrict__ ws, int M, int mode) {
  __shared__ float AT[32][KT + 4];
  __shared__ float BT[KT][Fh + 4];
  __shared__ float SROW[32];
  const int g    = blockIdx.x;
  const int row0 = blockIdx.y * 32;
  const int tid  = threadIdx.x;
  const int wave = tid >> 5, lane = tid & 31;
  const int half = lane >> 4, tl = lane & 15;
  const int cb   = wave * 16;

  float* arena = ws + (size_t)g * ARENA;
  const float* adjG   = adjBase + (size_t)g * adjGS;
  const float* XW     = arena + XW_OFF;
  const float* Hm     = arena + H_OFF;
  const float* DINV   = arena + DINV_OFF;
  const float* DEGINV = arena + DEGINV_OFF;
  const float* Bsrc   = (mode == 0) ? XW : Hm;

  float4 aReg[2], bReg[8];

  // prefetch chunk 0 into registers
#pragma unroll
  for (int p = 0; p < 2; ++p) {                       // A: 32x64 = 512 float4
    const int e = tid + 256 * p;
    const int r = e >> 4, c4 = (e & 15) << 2;
    aReg[p] = *(const float4*)&adjG[(size_t)(row0 + r) * M + c4];
  }
#pragma unroll
  for (int p = 0; p < 8; ++p) {                       // B: 64x128 = 2048 float4
    const int e = tid + 256 * p;
    const int kk = e >> 5, n4 = (e & 31) << 2;
    bReg[p] = *(const float4*)&Bsrc[(size_t)kk * Fh + n4];
  }

  v8f acc0 = {}, acc1 = {};
  for (int k0 = 0; k0 < M; k0 += KT) {
    __syncthreads();
    // commit registers -> LDS (b128 stores)
#pragma unroll
    for (int p = 0; p < 2; ++p) {
      const int e = tid + 256 * p;
      const int r = e >> 4, c4 = (e & 15) << 2;
      *(float4*)&AT[r][c4] = aReg[p];
    }
#pragma unroll
    for (int p = 0; p < 8; ++p) {
      const int e = tid + 256 * p;
      const int kk = e >> 5, n4 = (e & 31) << 2;
      float4 v = bReg[p];
      if (mode == 0) {
        const float sc = DINV[k0 + kk];
        v.x *= sc; v.y *= sc; v.z *= sc; v.w *= sc;
      }
      *(float4*)&BT[kk][n4] = v;
    }
    __syncthreads();

    // prefetch next chunk into registers while WMMAs consume LDS
    const int kn = k0 + KT;
    if (kn < M) {
#pragma unroll
      for (int p = 0; p < 2; ++p) {
        const int e = tid + 256 * p;
        const int r = e >> 4, c4 = (e & 15) << 2;
        aReg[p] = *(const float4*)&adjG[(size_t)(row0 + r) * M + kn + c4];
      }
#pragma unroll
      for (int p = 0; p < 8; ++p) {
        const int e = tid + 256 * p;
        const int kk = e >> 5, n4 = (e & 31) << 2;
        bReg[p] = *(const float4*)&Bsrc[(size_t)(kn + kk) * Fh + n4];
      }
    }

    for (int kk = 0; kk < KT; kk += 4) {
      v2f a0, a1, b;
      a0.x = AT[tl][kk + 2 * half];      a0.y = AT[tl][kk + 2 * half + 1];
      a1.x = AT[16 + tl][kk + 2 * half]; a1.y = AT[16 + tl][kk + 2 * half + 1];
      b.x  = BT[kk + 2 * half][cb + tl];
      b.y  = BT[kk + 2 * half + 1][cb + tl];
      acc0 = WMMA_F32(a0, b, acc0);
      acc1 = WMMA_F32(a1, b, acc1);
    }
  }

  if (mode == 0) {
    float* Hout = arena + H_OFF;
#pragma unroll
    for (int r = 0; r < 8; ++r) {
      const int i0 = row0 + r + 8 * half;
      const int i1 = i0 + 16;
      const float d0 = DINV[i0], d1 = DINV[i1];
      const float bn = bias[cb + tl];
      float v0 = d0 * acc0[r] + d0 * d0 * XW[(size_t)i0 * Fh + cb + tl] + bn;
      float v1 = d1 * acc1[r] + d1 * d1 * XW[(size_t)i1 * Fh + cb + tl] + bn;
      Hout[(size_t)i0 * Fh + cb + tl] = fmaxf(v0, 0.0f);
      Hout[(size_t)i1 * Fh + cb + tl] = fmaxf(v1, 0.0f);
    }
  } else {
    if (tid < 32) SROW[tid] = 0.0f;
    __syncthreads();
#pragma unroll
    for (int r = 0; r < 8; ++r) {
      const int i0 = row0 + r + 8 * half;
      const int i1 = i0 + 16;
      float v0 = fabsf(Hm[(size_t)i0 * Fh + cb + tl] - acc0[r] * DEGINV[i0]);
      float v1 = fabsf(Hm[(size_t)i1 * Fh + cb + tl] - acc1[r] * DEGINV[i1]);
      for (int m = 8; m >= 1; m >>= 1) {
        v0 += __shfl_xor(v0, m, 32);
        v1 += __shfl_xor(v1, m, 32);
      }
      if (tl == 0) {
        atomicAdd(&SROW[r + 8 * half], v0);        // ds_add_f32
        atomicAdd(&SROW[16 + r + 8 * half], v1);
      }
    }
    __syncthreads();
    if (tid < 32) (arena + S_OFF)[row0 + tid] = SROW[tid];
  }
}

// =====================================================================
// top-k: full in-LDS bitonic sort (desc value, asc index ties) -> idx[0..k)
// launched with blockDim == n (n is a power of two: 512 or 256)
// =====================================================================
__global__ void k_topk(float* __restrict__ ws, int n, int k) {
  __shared__ float key[512];
  __shared__ int   id[512];
  const int g = blockIdx.x, tid = threadIdx.x;
  float* arena = ws + (size_t)g * ARENA;
  if (tid < n) { key[tid] = (arena + S_OFF)[tid]; id[tid] = tid; }
  __syncthreads();
  for (int size = 2; size <= n; size <<= 1) {
    for (int stride = size >> 1; stride > 0; stride >>= 1) {
      const int i = tid, j = i ^ stride;
      if (j > i && j < n) {
        const float ki = key[i], kj = key[j];
        const int ii = id[i], ij = id[j];
        const bool before = (ki > kj) || (ki == kj && ii < ij);
        const bool up = ((i & size) == 0);
        if (up ? !before : before) { key[i] = kj; key[j] = ki; id[i] = ij; id[j] = ii; }
      }
      __syncthreads();
    }
  }
  if (tid < k) ((int*)(arena + IDX_OFF))[tid] = id[tid];
}

// =====================================================================
// gather pooled rows + attention source/dest dots (block = 128 = F)
// =====================================================================
__global__ void k_gather(const float* __restrict__ att_s,
                         const float* __restrict__ att_d,
                         float* __restrict__ ws, int k) {
  __shared__ float sb[128], db[128];
  const int g = blockIdx.x, i = blockIdx.y, f = threadIdx.x;
  float* arena = ws + (size_t)g * ARENA;
  const int src = ((const int*)(arena + IDX_OFF))[i];
  const float v = (arena + H_OFF)[(size_t)src * Fh + f];
  (arena + HP_OFF)[(size_t)i * Fh + f] = v;
  sb[f] = v * att_s[f];
  db[f] = v * att_d[f];
  __syncthreads();
  for (int s = 64; s >= 1; s >>= 1) {
    if (f < s) { sb[f] += sb[f + s]; db[f] += db[f + s]; }
    __syncthreads();
  }
  if (f == 0) { (arena + AS_OFF)[i] = sb[0]; (arena + AD_OFF)[i] = db[0]; }
}

// =====================================================================
// attn = softmax(leaky_relu(as_i + ad_j) + LAMB*adjPrev[idx_i, idx_j])
// block = 256 = 8 waves, wave per row; 3-pass softmax via LDS row buffer
// =====================================================================
__global__ void k_attn(const float* __restrict__ adjPrev, long adjGS, int ld,
                       float* __restrict__ ws, long aOutOff, int k) {
  __shared__ float rowbuf[8][256];
  const int g = blockIdx.x, tid = threadIdx.x;
  const int wave = tid >> 5, lane = tid & 31;
  const int i = blockIdx.y * 8 + wave;
  float* arena = ws + (size_t)g * ARENA;
  const int*   IDX = (const int*)(arena + IDX_OFF);
  const float* AD  = arena + AD_OFF;
  const float* adjG = adjPrev + (size_t)g * adjGS;
  const int   ii  = IDX[i];
  const float asi = (arena + AS_OFF)[i];

  float mx = -3.4e38f;
  for (int j = lane; j < k; j += 32) {
    float e = asi + AD[j];
    e = e > 0.0f ? e : NEG_SLOPE * e;
    e += LAMB * adjG[(size_t)ii * ld + IDX[j]];
    rowbuf[wave][j] = e;
    mx = fmaxf(mx, e);
  }
  for (int m = 16; m >= 1; m >>= 1) mx = fmaxf(mx, __shfl_xor(mx, m, 32));
  float sum = 0.0f;
  for (int j = lane; j < k; j += 32) sum += __expf(rowbuf[wave][j] - mx);
  for (int m = 16; m >= 1; m >>= 1) sum += __shfl_xor(sum, m, 32);
  const float inv = 1.0f / sum;
  float* Aout = arena + aOutOff;
  for (int j = lane; j < k; j += 32)
    Aout[(size_t)i * k + j] = __expf(rowbuf[wave][j] - mx) * inv;
}

// =====================================================================
// readout: [max_nodes | mean_nodes] -> 256 floats at dstOff
// =====================================================================
__global__ void k_readout(float* __restrict__ ws, long srcOff, int nodes, long dstOff) {
  const int g = blockIdx.x, f = threadIdx.x;
  const float* src = ws + (size_t)g * ARENA + srcOff;
  float mx = -3.4e38f, sm = 0.0f;
  for (int i = 0; i < nodes; ++i) {
    const float v = src[(size_t)i * Fh + f];
    mx = fmaxf(mx, v); sm += v;
  }
  float* dst = ws + (size_t)g * ARENA + dstOff;
  dst[f] = mx;
  dst[128 + f] = sm / (float)nodes;
}

// =====================================================================
// final MLP: z = relu(x1)+relu(x2)+relu(x3); 256->128->64->10; softmax
// =====================================================================
__global__ void k_mlp(const float* __restrict__ Wl1, const float* __restrict__ bl1,
                      const float* __restrict__ Wl2, const float* __restrict__ bl2,
                      const float* __restrict__ Wl3, const float* __restrict__ bl3,
                      const float* __restrict__ ws, float* __restrict__ out) {
  __shared__ float z[256], t1[128], t2[64], logits[Cc];
  const int g = blockIdx.x, tid = threadIdx.x;  // 128 threads
  const float* XR = ws + (size_t)g * ARENA + XR_OFF;
  for (int u = tid; u < 256; u += 128)
    z[u] = fmaxf(XR[u], 0.f) + fmaxf(XR[256 + u], 0.f) + fmaxf(XR[512 + u], 0.f);
  __syncthreads();
  {
    float acc = bl1[tid];
    for (int u = 0; u < 256; ++u) acc += z[u] * Wl1[(size_t)u * 128 + tid];
    t1[tid] = fmaxf(acc, 0.0f);
  }
  __syncthreads();
  if (tid < 64) {
    float acc = bl2[tid];
    for (int u = 0; u < 128; ++u) acc += t1[u] * Wl2[(size_t)u * 64 + tid];
    t2[tid] = fmaxf(acc, 0.0f);
  }
  __syncthreads();
  if (tid < Cc) {
    float acc = bl3[tid];
    for (int u = 0; u < 64; ++u) acc += t2[u] * Wl3[(size_t)u * Cc + tid];
    logits[tid] = acc;
  }
  __syncthreads();
  if (tid == 0) {
    float m = logits[0];
    for (int c = 1; c < Cc; ++c) m = fmaxf(m, logits[c]);
    float e[Cc], s = 0.0f;
    for (int c = 0; c < Cc; ++c) { e[c] = __expf(logits[c] - m); s += e[c]; }
    for (int c = 0; c < Cc; ++c) out[(size_t)g * Cc + c] = e[c] / s;
  }
}

// =====================================================================
extern "C" void kernel_launch(void* const* d_in, const int* in_sizes, int n_in,
                              void* d_out, int out_size, void* d_ws, size_t ws_size,
                              hipStream_t stream) {
  const float* x      = (const float*)d_in[0];
  const float* adj    = (const float*)d_in[1];
  const float* W1     = (const float*)d_in[2];
  const float* b1     = (const float*)d_in[3];
  const float* W2     = (const float*)d_in[4];
  const float* b2     = (const float*)d_in[5];
  const float* W3     = (const float*)d_in[6];
  const float* b3     = (const float*)d_in[7];
  const float* att1_s = (const float*)d_in[8];
  const float* att1_d = (const float*)d_in[9];
  const float* att2_s = (const float*)d_in[10];
  const float* att2_d = (const float*)d_in[11];
  const float* Wl1    = (const float*)d_in[12];
  const float* bl1    = (const float*)d_in[13];
  const float* Wl2    = (const float*)d_in[14];
  const float* bl2    = (const float*)d_in[15];
  const float* Wl3    = (const float*)d_in[16];
  const float* bl3    = (const float*)d_in[17];
  float* ws  = (float*)d_ws;
  float* out = (float*)d_out;

  const dim3 b256(256), b128(128);
  const long adjGS = (long)Nn * Nn;

  // ---- layer 1 (n = 512, adj = input adjacency) ----
  k_deg    <<<dim3(Gn, Nn / 8),  b256, 0, stream>>>(adj, adjGS, Nn, ws);
  k_xw     <<<dim3(Gn, Nn / 32), b256, 0, stream>>>(x, (long)Nn * Fh, W1, ws, Nn);
  k_agg    <<<dim3(Gn, Nn / 32), b256, 0, stream>>>(adj, adjGS, b1, ws, Nn, 0);
  k_agg    <<<dim3(Gn, Nn / 32), b256, 0, stream>>>(adj, adjGS, b1, ws, Nn, 1);
  k_topk   <<<Gn, Nn, 0, stream>>>(ws, Nn, Nn / 2);
  k_gather <<<dim3(Gn, Nn / 2),  b128, 0, stream>>>(att1_s, att1_d, ws, Nn / 2);
  k_attn   <<<dim3(Gn, Nn / 16), b256, 0, stream>>>(adj, adjGS, Nn, ws, A1_OFF, Nn / 2);
  k_readout<<<Gn, b128, 0, stream>>>(ws, HP_OFF, Nn / 2, XR_OFF);

  // ---- layer 2 (n = 256, adj = A1 in ws) ----
  const float* A1 = ws + A1_OFF;
  k_deg    <<<dim3(Gn, 32), b256, 0, stream>>>(A1, ARENA, 256, ws);
  k_xw     <<<dim3(Gn, 8),  b256, 0, stream>>>(ws + HP_OFF, ARENA, W2, ws, 256);
  k_agg    <<<dim3(Gn, 8),  b256, 0, stream>>>(A1, ARENA, b2, ws, 256, 0);
  k_agg    <<<dim3(Gn, 8),  b256, 0, stream>>>(A1, ARENA, b2, ws, 256, 1);
  k_topk   <<<Gn, 256, 0, stream>>>(ws, 256, 128);
  k_gather <<<dim3(Gn, 128), b128, 0, stream>>>(att2_s, att2_d, ws, 128);
  k_attn   <<<dim3(Gn, 16),  b256, 0, stream>>>(A1, ARENA, 256, ws, A2_OFF, 128);
  k_readout<<<Gn, b128, 0, stream>>>(ws, HP_OFF, 128, XR_OFF + 256);

  // ---- layer 3 (n = 128, adj = A2 in ws) ----
  const float* A2 = ws + A2_OFF;
  k_deg    <<<dim3(Gn, 16), b256, 0, stream>>>(A2, ARENA, 128, ws);
  k_xw     <<<dim3(Gn, 4),  b256, 0, stream>>>(ws + HP_OFF, ARENA, W3, ws, 128);
  k_agg    <<<dim3(Gn, 4),  b256, 0, stream>>>(A2, ARENA, b3, ws, 128, 0);
  k_readout<<<Gn, b128, 0, stream>>>(ws, H_OFF, 128, XR_OFF + 512);

  // ---- classifier head ----
  k_mlp<<<Gn, b128, 0, stream>>>(Wl1, bl1, Wl2, bl2, Wl3, bl3, ws, out);
}